// GraphAttentionLayer_Ind_91079076479130
// MI455X (gfx1250) — compile-verified
//
#include <hip/hip_runtime.h>
#include <math.h>

#define N_NODES 8192
#define F_IN    512
#define F_OUT   256
#define ALPHA   0.2f
#define NEG_BIG (-9.0e15f)
#define JT      64            // j-tile (K tile of attention GEMM)
#define PSTRIDE (JT + 5)      // 69: coprime w/ 64 banks -> conflict-free A-frag reads

typedef __attribute__((ext_vector_type(2))) float v2f;
typedef __attribute__((ext_vector_type(8))) float v8f;

// ---------------------------------------------------------------------------
// Kernel 1: H = X @ W   (fp32 WMMA 16x16x4)
// Block = 128 threads (4 waves), each wave one 16x16 tile -> block = 32x32.
// ---------------------------------------------------------------------------
__global__ __launch_bounds__(128) void k_gemm_xw(const float* __restrict__ X,
                                                 const float* __restrict__ W,
                                                 float* __restrict__ H) {
    const int lane  = threadIdx.x & 31;
    const int wave  = threadIdx.x >> 5;
    const int tileM = blockIdx.x * 32 + (wave >> 1) * 16;
    const int tileN = blockIdx.y * 32 + (wave & 1) * 16;
    const int l15   = lane & 15;
    const int half  = lane >> 4;   // 0: K pair {0,1}, 1: K pair {2,3}

    v8f acc = {};
    const float* xrow = X + (long long)(tileM + l15) * F_IN;
    for (int k = 0; k < F_IN; k += 4) {
        const int ka = k + half * 2;
        v2f a, b;
        a.x = xrow[ka];
        a.y = xrow[ka + 1];
        const float* wp = W + (long long)ka * F_OUT + tileN + l15;
        b.x = wp[0];
        b.y = wp[F_OUT];
        acc = __builtin_amdgcn_wmma_f32_16x16x4_f32(false, a, false, b,
                                                    (short)0, acc, false, false);
    }
    const int row0 = tileM + half * 8;
#pragma unroll
    for (int r = 0; r < 8; ++r)
        H[(long long)(row0 + r) * F_OUT + tileN + l15] = acc[r];
}

// ---------------------------------------------------------------------------
// Kernel 2: s_src = H @ a[:F], s_dst = H @ a[F:]   (one wave per row)
// ---------------------------------------------------------------------------
__global__ __launch_bounds__(256) void k_scores(const float* __restrict__ H,
                                                const float* __restrict__ a,
                                                float* __restrict__ s_src,
                                                float* __restrict__ s_dst) {
    const int lane = threadIdx.x & 31;
    const int row  = blockIdx.x * 8 + (threadIdx.x >> 5);
    float ss = 0.f, sd = 0.f;
    const float* hrow = H + (long long)row * F_OUT;
    for (int f = lane; f < F_OUT; f += 32) {
        const float hv = hrow[f];
        ss += hv * a[f];
        sd += hv * a[F_OUT + f];
    }
#pragma unroll
    for (int off = 16; off; off >>= 1) {
        ss += __shfl_xor(ss, off, 32);
        sd += __shfl_xor(sd, off, 32);
    }
    if (lane == 0) { s_src[row] = ss; s_dst[row] = sd; }
}

// ---------------------------------------------------------------------------
// Kernel 3: single-pass flash attention:  out = elu( softmax(att) @ H )
//   adj is streamed ONCE (256 MB) through double-buffered async LDS copies.
// Block = 256 threads (8 waves). Output tile: 16 rows x 256 cols.
// ---------------------------------------------------------------------------
__global__ __launch_bounds__(256) void k_flash_attn(const float* __restrict__ adj,
                                                    const float* __restrict__ H,
                                                    const float* __restrict__ s_src,
                                                    const float* __restrict__ s_dst,
                                                    float* __restrict__ out) {
    __shared__ float sdl[N_NODES];          // 32 KB: all dst scores
    __shared__ float rawA[2][16][JT];       // 8 KB: double-buffered adj tiles
    __shared__ float P[16][PSTRIDE];        // softmaxed attention tile
    __shared__ float pmax[16][17];          // per-(row, t16) tile max partials
    __shared__ float psum[16][17];          // per-(row, t16) tile sum partials
    __shared__ float mL[16], lL[16], scL[16], rsL[16];

    const int tid  = threadIdx.x;
    const int lane = tid & 31;
    const int wv   = tid >> 5;
    const int l15  = lane & 15;
    const int half = lane >> 4;
    const int r0   = blockIdx.x * 16;
    const int n0   = wv * 32;               // 8 waves x 32 cols = 256 cols

    // thread -> (row, 16-slot) mapping for the 16x64 tile passes
    const int tr = tid >> 4;                // 0..15 row in tile
    const int tt = tid & 15;                // 0..15 slot, 4 elements each

    // async-copy lane mapping: wave wv covers rows 2wv, 2wv+1 (one b128 per lane)
    const int cr = 2 * wv + half;           // tile row this lane copies
    const int cj = l15 * 4;                 // 4 floats per lane
    const uint32_t rawOff0 =
        (uint32_t)(uintptr_t)&rawA[0][cr][cj];
    const uint32_t rawOff1 =
        (uint32_t)(uintptr_t)&rawA[1][cr][cj];

    // preload scores
    for (int i = tid; i < N_NODES; i += 256) sdl[i] = s_dst[i];
    if (tid < 16) {
        rsL[tid] = s_src[r0 + tid];
        mL[tid]  = -INFINITY;
        lL[tid]  = 0.f;
    }
    psum[tr][tt] = 0.f;
    __syncthreads();

    // kick off async copy of tile 0
    {
        const float* gp = adj + (long long)(r0 + cr) * N_NODES + cj;
        asm volatile("global_load_async_to_lds_b128 %0, %1, off"
                     :: "v"(rawOff0), "v"(gp) : "memory");
    }

    v8f acc0 = {}, acc1 = {};
    const int NT = N_NODES / JT;
    for (int t = 0; t < NT; ++t) {
        const int buf = t & 1;
        const int j0  = t * JT;

        asm volatile("s_wait_asynccnt 0" ::: "memory");
        __syncthreads();                       // raw[buf] fully in LDS

        if (t + 1 < NT) {                      // overlap next tile's copy
            const float* gp = adj + (long long)(r0 + cr) * N_NODES + (j0 + JT) + cj;
            asm volatile("global_load_async_to_lds_b128 %0, %1, off"
                         :: "v"(buf ? rawOff0 : rawOff1), "v"(gp) : "memory");
        }

        // ---- pass A: att in registers + per-thread tile-max partial ----
        const float4 av = *(const float4*)&rawA[buf][tr][tt * 4];
        const float  si = rsL[tr];
        float att[4];
        {
            const float t0 = av.x, t1 = av.y, t2 = av.z, t3 = av.w;
            float e0 = si + sdl[j0 + tt * 4 + 0];
            float e1 = si + sdl[j0 + tt * 4 + 1];
            float e2 = si + sdl[j0 + tt * 4 + 2];
            float e3 = si + sdl[j0 + tt * 4 + 3];
            e0 = (e0 > 0.f) ? e0 : ALPHA * e0;
            e1 = (e1 > 0.f) ? e1 : ALPHA * e1;
            e2 = (e2 > 0.f) ? e2 : ALPHA * e2;
            e3 = (e3 > 0.f) ? e3 : ALPHA * e3;
            att[0] = (t0 > 0.f) ? e0 * t0 : NEG_BIG;
            att[1] = (t1 > 0.f) ? e1 * t1 : NEG_BIG;
            att[2] = (t2 > 0.f) ? e2 * t2 : NEG_BIG;
            att[3] = (t3 > 0.f) ? e3 * t3 : NEG_BIG;
        }
        pmax[tr][tt] = fmaxf(fmaxf(att[0], att[1]), fmaxf(att[2], att[3]));
        __syncthreads();

        // ---- pass B: fold previous tile's sums, new row max, scale ----
        if (tid < 16) {
            float tm = pmax[tid][0], ps = psum[tid][0];
#pragma unroll
            for (int q = 1; q < 16; ++q) {
                tm = fmaxf(tm, pmax[tid][q]);
                ps += psum[tid][q];
            }
            const float mo = mL[tid];
            const float nm = fmaxf(mo, tm);
            const float sc = __expf(mo - nm);
            scL[tid] = sc;
            mL[tid]  = nm;
            lL[tid]  = (lL[tid] + ps) * sc;
        }
        __syncthreads();

        // ---- pass C: p = exp(att - m) into P, sum partials, rescale acc ----
        {
            const float nm = mL[tr];
            const float p0 = __expf(att[0] - nm);
            const float p1 = __expf(att[1] - nm);
            const float p2 = __expf(att[2] - nm);
            const float p3 = __expf(att[3] - nm);
            P[tr][tt * 4 + 0] = p0;
            P[tr][tt * 4 + 1] = p1;
            P[tr][tt * 4 + 2] = p2;
            P[tr][tt * 4 + 3] = p3;
            psum[tr][tt] = (p0 + p1) + (p2 + p3);
        }
#pragma unroll
        for (int r = 0; r < 8; ++r) {
            const float sc = scL[half * 8 + r];
            acc0[r] *= sc;
            acc1[r] *= sc;
        }
        __syncthreads();

        // ---- pass D: 16 K-steps of fp32 WMMA over this tile ----
#pragma unroll
        for (int kk = 0; kk < JT; kk += 4) {
            v2f a, b0, b1;
            a.x = P[l15][kk + half * 2];
            a.y = P[l15][kk + half * 2 + 1];
            const float* hb = H + (long long)(j0 + kk + half * 2) * F_OUT;
            b0.x = hb[n0 + l15];
            b0.y = hb[F_OUT + n0 + l15];
            b1.x = hb[n0 + 16 + l15];
            b1.y = hb[F_OUT + n0 + 16 + l15];
            acc0 = __builtin_amdgcn_wmma_f32_16x16x4_f32(false, a, false, b0,
                                                         (short)0, acc0, false, false);
            acc1 = __builtin_amdgcn_wmma_f32_16x16x4_f32(false, a, false, b1,
                                                         (short)0, acc1, false, false);
        }
        __syncthreads();                       // protect P/pmax before next tile
    }

    // ---- epilogue: fold last tile's sums, normalize, ELU, store ----
    if (tid < 16) {
        float ps = psum[tid][0];
#pragma unroll
        for (int q = 1; q < 16; ++q) ps += psum[tid][q];
        lL[tid] += ps;
    }
    __syncthreads();

    const int row0 = r0 + half * 8;
#pragma unroll
    for (int r = 0; r < 8; ++r) {
        const float inv = 1.f / lL[half * 8 + r];
        float v0 = acc0[r] * inv, v1 = acc1[r] * inv;
        v0 = (v0 > 0.f) ? v0 : expm1f(v0);     // ELU (alpha=1)
        v1 = (v1 > 0.f) ? v1 : expm1f(v1);
        out[(long long)(row0 + r) * F_OUT + n0 + l15]      = v0;
        out[(long long)(row0 + r) * F_OUT + n0 + 16 + l15] = v1;
    }
}

// ---------------------------------------------------------------------------
extern "C" void kernel_launch(void* const* d_in, const int* in_sizes, int n_in,
                              void* d_out, int out_size, void* d_ws, size_t ws_size,
                              hipStream_t stream) {
    const float* X   = (const float*)d_in[0];
    const float* adj = (const float*)d_in[1];
    const float* W   = (const float*)d_in[2];
    const float* a   = (const float*)d_in[3];
    float* out = (float*)d_out;

    float* H     = (float*)d_ws;                 // N x F_OUT  (8 MB)
    float* s_src = H + (size_t)N_NODES * F_OUT;  // N
    float* s_dst = s_src + N_NODES;              // N

    dim3 g1(N_NODES / 32, F_OUT / 32);
    k_gemm_xw<<<g1, 128, 0, stream>>>(X, W, H);
    k_scores<<<N_NODES / 8, 256, 0, stream>>>(H, a, s_src, s_dst);
    k_flash_attn<<<N_NODES / 16, 256, 0, stream>>>(adj, H, s_src, s_dst, out);
}